// GATLayer_37211596652679
// MI455X (gfx1250) — compile-verified
//
#include <hip/hip_runtime.h>
#include <hip/hip_bf16.h>

#define N_NODES  100000
#define N_EDGES  1600000
#define IN_FEAT  128
#define N_HEADS  8
#define OUT_FEAT 16
#define HF       128
#define NEG_SLOPE 0.2f

typedef float v2f __attribute__((ext_vector_type(2)));
typedef float v8f __attribute__((ext_vector_type(8)));

// ---------------------------------------------------------------------------
// Kernel 1: h = x @ Wv + bv   via V_WMMA_F32_16X16X4_F32
// One wave computes a 16-row x 128-col strip of h. 8 accumulator tiles.
// N_NODES = 100000 = 6250 * 16 -> no row tail; strip guard is wave-uniform
// so EXEC is all-ones for every WMMA (ISA requirement).
// ---------------------------------------------------------------------------
__global__ __launch_bounds__(256) void gemm_h_kernel(
    const float* __restrict__ x, const float* __restrict__ Wv,
    const float* __restrict__ bv, float* __restrict__ h)
{
    const int wave = threadIdx.x >> 5;
    const int lane = threadIdx.x & 31;
    const int strip = blockIdx.x * 8 + wave;
    if (strip >= N_NODES / 16) return;           // wave-uniform exit
    const int m0 = strip * 16;

    const int lane15 = lane & 15;
    const int kofs   = (lane >> 4) * 2;          // lanes 0-15: K+0,1 ; 16-31: K+2,3
    const float* xrow = x + (size_t)(m0 + lane15) * IN_FEAT + kofs;

    v8f acc[8];
    #pragma unroll
    for (int t = 0; t < 8; ++t) acc[t] = (v8f){0.f,0.f,0.f,0.f,0.f,0.f,0.f,0.f};

    for (int kk = 0; kk < IN_FEAT; kk += 4) {
        const v2f a = *(const v2f*)(xrow + kk);                  // 8B aligned
        const float* wp = Wv + (size_t)(kk + kofs) * HF + lane15;
        #pragma unroll
        for (int t = 0; t < 8; ++t) {
            v2f b;
            b.x = wp[t * 16];
            b.y = wp[t * 16 + HF];
            acc[t] = __builtin_amdgcn_wmma_f32_16x16x4_f32(
                false, a, false, b, (short)0, acc[t], false, false);
        }
    }

    // D layout: elem r -> row M = r + (lane<16 ? 0 : 8), col N = lane&15
    const int rowAdd = (lane >> 4) << 3;
    #pragma unroll
    for (int t = 0; t < 8; ++t) {
        const int col = t * 16 + lane15;
        const float bias = bv[col];
        #pragma unroll
        for (int r = 0; r < 8; ++r) {
            h[(size_t)(m0 + r + rowAdd) * HF + col] = acc[t][r] + bias;
        }
    }
}

// ---------------------------------------------------------------------------
// Kernel 2: per-node q = h@Wq+bq, k = h@Wk+bk; init smax/denom; zero vout.
// One wave per node; lane holds 4 features of the h row.
// ---------------------------------------------------------------------------
__global__ __launch_bounds__(256) void qk_init_kernel(
    const float* __restrict__ h,
    const float* __restrict__ Wq, const float* __restrict__ bq,
    const float* __restrict__ Wk, const float* __restrict__ bk,
    float* __restrict__ q, float* __restrict__ kf,
    float* __restrict__ smax, float* __restrict__ denom,
    float* __restrict__ vout)
{
    const int node = blockIdx.x * 8 + (threadIdx.x >> 5);
    if (node >= N_NODES) return;                 // wave-uniform
    const int lane = threadIdx.x & 31;

    const float4 hv = *(const float4*)(h + (size_t)node * HF + lane * 4);
    *(float4*)(vout + (size_t)node * HF + lane * 4) = make_float4(0.f, 0.f, 0.f, 0.f);

    const int j = lane * 4;
    #pragma unroll
    for (int hd = 0; hd < N_HEADS; ++hd) {
        float pq = hv.x * Wq[(j + 0) * N_HEADS + hd] + hv.y * Wq[(j + 1) * N_HEADS + hd]
                 + hv.z * Wq[(j + 2) * N_HEADS + hd] + hv.w * Wq[(j + 3) * N_HEADS + hd];
        float pk = hv.x * Wk[(j + 0) * N_HEADS + hd] + hv.y * Wk[(j + 1) * N_HEADS + hd]
                 + hv.z * Wk[(j + 2) * N_HEADS + hd] + hv.w * Wk[(j + 3) * N_HEADS + hd];
        #pragma unroll
        for (int off = 16; off > 0; off >>= 1) {
            pq += __shfl_xor(pq, off, 32);
            pk += __shfl_xor(pk, off, 32);
        }
        if (lane == 0) {
            q [node * N_HEADS + hd] = pq + bq[hd];
            kf[node * N_HEADS + hd] = pk + bk[hd];
            smax [node * N_HEADS + hd] = -1e30f;
            denom[node * N_HEADS + hd] = 0.f;
        }
    }
}

// Float atomic max via sign-split int/uint atomics (IEEE-correct, mixed sign).
__device__ __forceinline__ void atomicMaxF(float* addr, float v) {
    if (v >= 0.f) atomicMax((int*)addr, __float_as_int(v));
    else          atomicMin((unsigned int*)addr, __float_as_uint(v));
}

// ---------------------------------------------------------------------------
// Kernel 3: per-edge score + leaky relu, store coeff, segment max into smax.
// ---------------------------------------------------------------------------
__global__ __launch_bounds__(256) void edge_max_kernel(
    const int* __restrict__ src, const int* __restrict__ dst,
    const float* __restrict__ q, const float* __restrict__ kf,
    float* __restrict__ coeff, float* __restrict__ smax)
{
    const int e = blockIdx.x * blockDim.x + threadIdx.x;
    if (e >= N_EDGES) return;
    const int s = src[e], d = dst[e];
    #pragma unroll
    for (int hd = 0; hd < N_HEADS; ++hd) {
        float c = kf[s * N_HEADS + hd] + q[d * N_HEADS + hd];
        c = (c >= 0.f) ? c : NEG_SLOPE * c;
        coeff[(size_t)e * N_HEADS + hd] = c;
        atomicMaxF(&smax[d * N_HEADS + hd], c);
    }
}

// ---------------------------------------------------------------------------
// Kernel 4: ex = exp(c - smax[dst]); store in-place; denom += ex.
// ---------------------------------------------------------------------------
__global__ __launch_bounds__(256) void edge_sum_kernel(
    const int* __restrict__ dst, float* __restrict__ coeff,
    const float* __restrict__ smax, float* __restrict__ denom)
{
    const int e = blockIdx.x * blockDim.x + threadIdx.x;
    if (e >= N_EDGES) return;
    const int d = dst[e];
    #pragma unroll
    for (int hd = 0; hd < N_HEADS; ++hd) {
        const float ex = __expf(coeff[(size_t)e * N_HEADS + hd] - smax[d * N_HEADS + hd]);
        coeff[(size_t)e * N_HEADS + hd] = ex;
        atomicAdd(&denom[d * N_HEADS + hd], ex);
    }
}

// ---------------------------------------------------------------------------
// Kernel 5: weighted message scatter: vout[dst] += h[src] * attn.
// One wave per edge; lane covers 4 consecutive features (one head / 4 lanes).
// Atomic targets are L2-resident (vout = 51 MB < 192 MB L2).
// ---------------------------------------------------------------------------
__global__ __launch_bounds__(256) void edge_agg_kernel(
    const int* __restrict__ src, const int* __restrict__ dst,
    const float* __restrict__ h, const float* __restrict__ ex,
    const float* __restrict__ denom, float* __restrict__ vout)
{
    const int e = blockIdx.x * 8 + (threadIdx.x >> 5);
    if (e >= N_EDGES) return;                    // wave-uniform
    const int lane = threadIdx.x & 31;
    const int s = src[e], d = dst[e];
    const int hd = lane >> 2;                    // (lane*4)/OUT_FEAT
    const float attn = ex[(size_t)e * N_HEADS + hd] / denom[d * N_HEADS + hd];

    const float4 u4 = *(const float4*)(h + (size_t)s * HF + lane * 4);
    float* vp = vout + (size_t)d * HF + lane * 4;
    atomicAdd(vp + 0, u4.x * attn);
    atomicAdd(vp + 1, u4.y * attn);
    atomicAdd(vp + 2, u4.z * attn);
    atomicAdd(vp + 3, u4.w * attn);
}

// ---------------------------------------------------------------------------
// Kernel 6: out[n,f] = mean over heads of vout[n, hd*16 + f].
// ---------------------------------------------------------------------------
__global__ __launch_bounds__(256) void mean_kernel(
    const float* __restrict__ vout, float* __restrict__ out)
{
    const int idx = blockIdx.x * blockDim.x + threadIdx.x;
    if (idx >= N_NODES * OUT_FEAT) return;
    const int n = idx >> 4;
    const int f = idx & 15;
    float sum = 0.f;
    #pragma unroll
    for (int hd = 0; hd < N_HEADS; ++hd)
        sum += vout[(size_t)n * HF + hd * OUT_FEAT + f];
    out[idx] = sum * (1.f / N_HEADS);
}

extern "C" void kernel_launch(void* const* d_in, const int* in_sizes, int n_in,
                              void* d_out, int out_size, void* d_ws, size_t ws_size,
                              hipStream_t stream) {
    const float* x   = (const float*)d_in[0];
    const int*   src = (const int*)  d_in[1];
    const int*   dst = (const int*)  d_in[2];
    const float* Wv  = (const float*)d_in[3];
    const float* bv  = (const float*)d_in[4];
    const float* Wq  = (const float*)d_in[5];
    const float* bq  = (const float*)d_in[6];
    const float* Wk  = (const float*)d_in[7];
    const float* bk  = (const float*)d_in[8];
    float* out = (float*)d_out;

    // Workspace carve-up (all 16B aligned slices).
    char* ws = (char*)d_ws;
    float* h     = (float*)ws;  ws += (size_t)N_NODES * HF      * sizeof(float); // 51.2 MB
    float* q     = (float*)ws;  ws += (size_t)N_NODES * N_HEADS * sizeof(float); //  3.2 MB
    float* kf    = (float*)ws;  ws += (size_t)N_NODES * N_HEADS * sizeof(float);
    float* smax  = (float*)ws;  ws += (size_t)N_NODES * N_HEADS * sizeof(float);
    float* denom = (float*)ws;  ws += (size_t)N_NODES * N_HEADS * sizeof(float);
    float* coeff = (float*)ws;  ws += (size_t)N_EDGES * N_HEADS * sizeof(float); // 51.2 MB
    float* vout  = (float*)ws;  ws += (size_t)N_NODES * HF      * sizeof(float); // 51.2 MB

    const int strips = N_NODES / 16;                       // 6250
    gemm_h_kernel<<<(strips + 7) / 8, 256, 0, stream>>>(x, Wv, bv, h);

    qk_init_kernel<<<(N_NODES + 7) / 8, 256, 0, stream>>>(h, Wq, bq, Wk, bk,
                                                          q, kf, smax, denom, vout);

    edge_max_kernel<<<(N_EDGES + 255) / 256, 256, 0, stream>>>(src, dst, q, kf, coeff, smax);

    edge_sum_kernel<<<(N_EDGES + 255) / 256, 256, 0, stream>>>(dst, coeff, smax, denom);

    edge_agg_kernel<<<(N_EDGES + 7) / 8, 256, 0, stream>>>(src, dst, h, coeff, denom, vout);

    mean_kernel<<<(N_NODES * OUT_FEAT + 255) / 256, 256, 0, stream>>>(vout, out);
}